// Fca_SE_GatingModule_70007966925059
// MI455X (gfx1250) — compile-verified
//
#include <hip/hip_runtime.h>
#include <hip/hip_bf16.h>
#include <math.h>

// ---------------------------------------------------------------------------
// Problem constants (from reference): x[32,512,56,56] fp32, C=512, HW=3136
// Outputs concatenated: out(51380224) | bounded(16384) | raw(16384) |
//                       mask(16384)   | squeezed(16384)
// ---------------------------------------------------------------------------
#define BATCH   32
#define CHAN    512
#define HW      3136          // 56*56
#define HW4     784           // HW / 4
#define HID     128           // CHAN / 4

typedef __attribute__((ext_vector_type(2))) float v2f;
typedef __attribute__((ext_vector_type(4))) float v4f;
typedef __attribute__((ext_vector_type(8))) float v8f;

// ---------------------------------------------------------------------------
// Kernel 1: multi-spectral DCT squeeze.  One 256-thread block per (b,c) row.
// squeezed[b,c] = sum_s x[b,c,s] * dct[c,s]
// ---------------------------------------------------------------------------
__global__ __launch_bounds__(256) void squeeze_kernel(
    const float* __restrict__ x, const float* __restrict__ dct,
    float* __restrict__ squeezed) {
  const int bc = blockIdx.x;                 // 0 .. BATCH*CHAN-1
  const int c  = bc & (CHAN - 1);
  const v4f* __restrict__ xr = (const v4f*)(x   + (size_t)bc * HW);
  const v4f* __restrict__ dr = (const v4f*)(dct + (size_t)c  * HW);

  float acc = 0.0f;
  for (int i = threadIdx.x; i < HW4; i += 256) {
    v4f xv = xr[i];
    v4f dv = dr[i];
    acc += xv.x * dv.x + xv.y * dv.y + xv.z * dv.z + xv.w * dv.w;
  }
  // wave32 reduction, then cross-wave via LDS
  for (int off = 16; off > 0; off >>= 1)
    acc += __shfl_down(acc, off, 32);
  __shared__ float part[8];
  if ((threadIdx.x & 31) == 0) part[threadIdx.x >> 5] = acc;
  __syncthreads();
  if (threadIdx.x == 0) {
    float s = 0.0f;
    for (int w = 0; w < 8; ++w) s += part[w];
    squeezed[bc] = s;
  }
}

// ---------------------------------------------------------------------------
// Kernel 2: excitation MLP with fp32 WMMA (V_WMMA_F32_16X16X4_F32).
// One block, 512 threads = 16 waves.
//   GEMM1: [32,512] x W1^T[512,128] -> relu -> hid (LDS)
//   GEMM2: [32,128] x W2^T[128,512] -> raw, bounded = tanh(raw)
// A 16x4 f32 layout : lanes 0-15 M=lid hold K=k,k+1 ; lanes 16-31 K=k+2,k+3
// B 4x16 f32 layout : lanes 0-15 N=lid hold K=k,k+1 ; lanes 16-31 K=k+2,k+3
// C/D 16x16 f32     : VGPR r -> row m0+r (lanes 0-15) / m0+r+8 (lanes 16-31)
// ---------------------------------------------------------------------------
__global__ __launch_bounds__(512) void mlp_kernel(
    const float* __restrict__ sq, const float* __restrict__ W1,
    const float* __restrict__ W2, float* __restrict__ raw_out,
    float* __restrict__ bounded_out) {
  __shared__ float hid[BATCH * HID];          // 16 KB

  const int lane = threadIdx.x & 31;
  const int wave = threadIdx.x >> 5;          // 0..15
  const int half = lane >> 4;                 // 0: K=k,k+1   1: K=k+2,k+3
  const int lid  = lane & 15;

  // ---- GEMM1: 2 x 8 tiles of 16x16, one tile per wave, K = 512 ----
  {
    const int m0 = (wave >> 3) * 16;          // 0 or 16
    const int n0 = (wave & 7) * 16;           // 0..112
    const float* __restrict__ arow = sq + (size_t)(m0 + lid) * CHAN;
    const float* __restrict__ brow = W1 + (size_t)(n0 + lid) * CHAN; // B[k][n]=W1[n][k]
    v8f acc = {};
    for (int k = 0; k < CHAN; k += 4) {
      const int k0 = k + 2 * half;
      v2f a; a.x = arow[k0]; a.y = arow[k0 + 1];
      v2f b; b.x = brow[k0]; b.y = brow[k0 + 1];
      acc = __builtin_amdgcn_wmma_f32_16x16x4_f32(
          false, a, false, b, (short)0, acc, false, false);
    }
    for (int r = 0; r < 8; ++r) {
      const int m = m0 + r + 8 * half;
      const float v = acc[r];
      hid[m * HID + n0 + lid] = v > 0.0f ? v : 0.0f;   // ReLU
    }
  }
  __syncthreads();

  // ---- GEMM2: 2 x 32 tiles of 16x16, 4 tiles per wave, K = 128 ----
  for (int t = 0; t < 4; ++t) {
    const int tile = wave * 4 + t;            // 0..63
    const int m0 = (tile >> 5) * 16;          // 0 or 16
    const int n0 = (tile & 31) * 16;          // 0..496
    const float* __restrict__ arow = hid + (size_t)(m0 + lid) * HID;
    const float* __restrict__ brow = W2 + (size_t)(n0 + lid) * HID;  // B[k][n]=W2[n][k]
    v8f acc = {};
    for (int k = 0; k < HID; k += 4) {
      const int k0 = k + 2 * half;
      v2f a; a.x = arow[k0]; a.y = arow[k0 + 1];
      v2f b; b.x = brow[k0]; b.y = brow[k0 + 1];
      acc = __builtin_amdgcn_wmma_f32_16x16x4_f32(
          false, a, false, b, (short)0, acc, false, false);
    }
    for (int r = 0; r < 8; ++r) {
      const int m = m0 + r + 8 * half;
      const int n = n0 + lid;
      const float v = acc[r];
      raw_out[m * CHAN + n]     = v;
      bounded_out[m * CHAN + n] = tanhf(v);
    }
  }
}

// ---------------------------------------------------------------------------
// Kernel 3: stable descending rank -> binary top-k mask.
// rank[c] = #{ c' : raw[c'] > raw[c]  ||  (raw[c'] == raw[c] && c' < c) }
// (exactly JAX's argsort(argsort(-raw)) with stable tie-breaking)
// ---------------------------------------------------------------------------
__global__ __launch_bounds__(256) void gate_kernel(
    const float* __restrict__ raw, const int* __restrict__ ktab,
    float* __restrict__ mask_out) {
  const int b = blockIdx.x;
  __shared__ float rs[CHAN];
  for (int i = threadIdx.x; i < CHAN; i += 256) rs[i] = raw[b * CHAN + i];
  __syncthreads();
  const int k = ktab[b];
  for (int c = threadIdx.x; c < CHAN; c += 256) {
    const float v = rs[c];
    int cnt = 0;
    for (int j = 0; j < CHAN; ++j) {
      const float u = rs[j];
      cnt += (u > v) || (u == v && j < c);
    }
    mask_out[b * CHAN + c] = (cnt < k) ? 1.0f : 0.0f;
  }
}

// ---------------------------------------------------------------------------
// Kernel 4: out = x * mask[b,c].  x loads use regular (RT) policy — with the
// non-temporal stores below, most of x (205.5 MB vs 192 MB L2) stays L2
// resident from kernel 1.  out is written once and never re-read: NT stores.
// ---------------------------------------------------------------------------
__global__ __launch_bounds__(256) void apply_kernel(
    const float* __restrict__ x, const float* __restrict__ mask,
    float* __restrict__ out) {
  const int bc = blockIdx.x;
  const float g = mask[bc];
  const v4f* __restrict__ xr = (const v4f*)(x + (size_t)bc * HW);
  v4f* __restrict__ orow = (v4f*)(out + (size_t)bc * HW);
  for (int i = threadIdx.x; i < HW4; i += 256) {
    v4f v = xr[i];
    v.x *= g; v.y *= g; v.z *= g; v.w *= g;
    __builtin_nontemporal_store(v, orow + i);
  }
}

// ---------------------------------------------------------------------------
// Launch
// ---------------------------------------------------------------------------
extern "C" void kernel_launch(void* const* d_in, const int* in_sizes, int n_in,
                              void* d_out, int out_size, void* d_ws, size_t ws_size,
                              hipStream_t stream) {
  (void)in_sizes; (void)n_in; (void)out_size; (void)d_ws; (void)ws_size;
  const float* x    = (const float*)d_in[0];
  const float* W1   = (const float*)d_in[1];
  const float* W2   = (const float*)d_in[2];
  const float* dct  = (const float*)d_in[3];
  const int*   ktab = (const int*)d_in[4];

  float* out = (float*)d_out;
  const size_t N0   = (size_t)BATCH * CHAN * HW;   // 51,380,224
  float* bounded  = out + N0;
  float* raw      = bounded + (size_t)BATCH * CHAN;
  float* mask     = raw + (size_t)BATCH * CHAN;
  float* squeezed = mask + (size_t)BATCH * CHAN;

  squeeze_kernel<<<BATCH * CHAN, 256, 0, stream>>>(x, dct, squeezed);
  mlp_kernel<<<1, 512, 0, stream>>>(squeezed, W1, W2, raw, bounded);
  gate_kernel<<<BATCH, 256, 0, stream>>>(raw, ktab, mask);
  apply_kernel<<<BATCH * CHAN, 256, 0, stream>>>(x, mask, out);
}